// ICGP_Convnp_19696720019898
// MI455X (gfx1250) — compile-verified
//
#include <hip/hip_runtime.h>
#include <hip/hip_bf16.h>

typedef float v2f __attribute__((ext_vector_type(2)));
typedef float v8f __attribute__((ext_vector_type(8)));

#define N1   1024
#define N2   1024
#define KC   15      // K*C = 5*3
#define NB   8       // batch

// out[b,m,k,c] = sum_n f[b,n,k,c] * exp(-0.5*((xa[b,n,c]-xt[b,m,c])/s[k,c])^2)
// Rewritten base-2:  w = exp2( -(r*(xa - xt))^2 ),  r = sqrt(0.5*log2 e)/s.
//
// Per block: fixed (b, kc), 128 m-values (8 waves x 16 rows).
// Per wave:  16 m rows; loop n in chunks of 4 feeding V_WMMA_F32_16X16X4_F32:
//   A[m,kk] = w(m, n0+kk)   (computed in-register: 2 exps/lane)
//   B[kk,j] = f[n0+kk]      (same value every column j)
//   C      += A*B           -> row sums replicated across the 16 columns.
__global__ __launch_bounds__(256)
void icgp_rbf_smooth_wmma(const float* __restrict__ xa,
                          const float* __restrict__ features,
                          const float* __restrict__ xt,
                          const float* __restrict__ log_std,
                          float* __restrict__ out)
{
    __shared__ float2 sp[N1];              // interleaved (a' = xa*r, f) pairs: 8 KB

    const int b   = blockIdx.z;
    const int kc  = blockIdx.y;            // kc = k*3 + c
    const int c   = kc % 3;
    const int tid = threadIdx.x;

    // r = sqrt(0.5 * log2(e)) / (exp(log_std)+eps)
    const float s = __builtin_amdgcn_exp2f(log_std[kc] * 1.4426950408889634f) + 1e-6f;
    const float r = 0.84961580047383f / s; // sqrt(0.72134752044448)

    // Stage scaled context points + features into LDS (whole n-range for this b,kc).
    for (int n = tid; n < N1; n += 256) {
        float a = xa[(b * N1 + n) * 3 + c] * r;
        float f = features[(b * N1 + n) * KC + kc];
        sp[n] = make_float2(a, f);
    }
    __syncthreads();

    const int lane = tid & 31;
    const int wave = tid >> 5;
    const int j    = lane & 15;            // A-row index within tile / B column
    const int hi   = lane >> 4;            // 0: K=0,1   1: K=2,3
    const int m    = blockIdx.x * 128 + wave * 16 + j;

    const float t = xt[(b * N2 + m) * 3 + c] * r;

    v8f acc0 = {};
    v8f acc1 = {};
    for (int n0 = 0; n0 < N1; n0 += 8) {
        // ---- chunk 0: n0 .. n0+3 ----
        {
            const int na = n0 + 2 * hi;
            float2 p0 = sp[na];            // (a', f) — 16B-aligned pair load, LDS broadcast
            float2 p1 = sp[na + 1];
            float d0 = p0.x - t;
            float d1 = p1.x - t;
            float w0 = __builtin_amdgcn_exp2f(-d0 * d0);
            float w1 = __builtin_amdgcn_exp2f(-d1 * d1);
            v2f A  = { w0, w1 };           // A[m, K=2hi], A[m, K=2hi+1]
            v2f Bv = { p0.y, p1.y };       // B[K=2hi, j], B[K=2hi+1, j]
            acc0 = __builtin_amdgcn_wmma_f32_16x16x4_f32(
                       false, A, false, Bv, (short)0, acc0, false, false);
        }
        // ---- chunk 1: n0+4 .. n0+7 (independent accumulator for ILP) ----
        {
            const int na = n0 + 4 + 2 * hi;
            float2 p0 = sp[na];
            float2 p1 = sp[na + 1];
            float d0 = p0.x - t;
            float d1 = p1.x - t;
            float w0 = __builtin_amdgcn_exp2f(-d0 * d0);
            float w1 = __builtin_amdgcn_exp2f(-d1 * d1);
            v2f A  = { w0, w1 };
            v2f Bv = { p0.y, p1.y };
            acc1 = __builtin_amdgcn_wmma_f32_16x16x4_f32(
                       false, A, false, Bv, (short)0, acc1, false, false);
        }
    }

    // C/D layout: VGPR g holds row M=g (lanes 0-15) and M=g+8 (lanes 16-31),
    // and every column carries the same row-sum. Lane j==0 of each half owns
    // 8 complete rows in acc[0..7].
    if (j == 0) {
        const int mrow0 = blockIdx.x * 128 + wave * 16 + hi * 8;
        v8f acc = acc0 + acc1;
        #pragma unroll
        for (int rr = 0; rr < 8; ++rr) {
            out[(b * N2 + (mrow0 + rr)) * KC + kc] = acc[rr];
        }
    }
}

extern "C" void kernel_launch(void* const* d_in, const int* in_sizes, int n_in,
                              void* d_out, int out_size, void* d_ws, size_t ws_size,
                              hipStream_t stream) {
    const float* xa       = (const float*)d_in[0];
    const float* features = (const float*)d_in[1];
    const float* xt       = (const float*)d_in[2];
    const float* log_std  = (const float*)d_in[3];
    float* out            = (float*)d_out;

    dim3 grid(N2 / 128, KC, NB);   // (m-tiles, k*c, batch) = (8, 15, 8)
    icgp_rbf_smooth_wmma<<<grid, 256, 0, stream>>>(xa, features, xt, log_std, out);
}